// PatNetv1_9680856285698
// MI455X (gfx1250) — compile-verified
//
#include <hip/hip_runtime.h>
#include <hip/hip_bf16.h>

#define BATCH 4096
#define HW    225      // 15*15
#define HWP   240      // padded to 15 tiles of 16
#define D     48
#define P     16
#define V     32
#define PCODE 2380

typedef __attribute__((ext_vector_type(2))) float v2f;
typedef __attribute__((ext_vector_type(8))) float v8f;

__global__ __launch_bounds__(256) void patnet_fused(
    const float* __restrict__ emb,
    const float* __restrict__ conv_w, const float* __restrict__ conv_b,
    const float* __restrict__ ppw_w,  const float* __restrict__ ppw_b,
    const float* __restrict__ pdw_w,  const float* __restrict__ pdw_b,
    const float* __restrict__ pf_w,
    const float* __restrict__ vpw_w,  const float* __restrict__ vpw_b,
    const float* __restrict__ vl1_w,  const float* __restrict__ vl1_b,
    const float* __restrict__ vl2_w,  const float* __restrict__ vl2_b,
    const float* __restrict__ vf_w,   const float* __restrict__ vf_b,
    const int* __restrict__ sparse,   const int* __restrict__ board,
    float* __restrict__ out_val,      float* __restrict__ out_pol)
{
    __shared__ float s_feat0[D * HW];    // embedding sums, [ch][px]
    __shared__ float s_feat [D * HWP];   // post depthwise conv, [ch][px], zero-padded px 225..239
    __shared__ float s_pol1 [P * HWP];   // policy pointwise output, [ch][px] (pad region never read)
    __shared__ float s_vsum [V];         // value channel sums

    const int img = blockIdx.x;
    const int tid = threadIdx.x;
    const int px  = tid;

    // ---------------- Stage 1: embedding gather + sum ----------------
    if (tid < V) s_vsum[tid] = 0.0f;
    if (px < HW) {
        const int* sp = sparse + (size_t)img * 12 * HW;
        const int* bd = board  + (size_t)img * 2  * HW;
        int c0 = (bd[px]      > 0) ? PCODE : sp[10 * HW + px];
        int c1 = (bd[HW + px] > 0) ? PCODE : sp[11 * HW + px];
        c1 += PCODE + 1;
        const float4* e0 = (const float4*)emb + c0 * (D / 4);
        const float4* e1 = (const float4*)emb + c1 * (D / 4);
        #pragma unroll
        for (int j = 0; j < D / 4; ++j) {
            float4 a = e0[j], b = e1[j];
            s_feat0[(4 * j + 0) * HW + px] = a.x + b.x;
            s_feat0[(4 * j + 1) * HW + px] = a.y + b.y;
            s_feat0[(4 * j + 2) * HW + px] = a.z + b.z;
            s_feat0[(4 * j + 3) * HW + px] = a.w + b.w;
        }
    }
    __syncthreads();

    // ---------------- Stage 2: depthwise 3x3 conv (48 ch) + ReLU ----------------
    if (px < HW) {
        const int y = px / 15, x = px % 15;
        for (int ch = 0; ch < D; ++ch) {
            float acc = conv_b[ch];
            const float* w  = conv_w + ch * 9;
            const float* f0 = s_feat0 + ch * HW;
            #pragma unroll
            for (int ky = 0; ky < 3; ++ky) {
                const int iy = y + ky - 1;
                if ((unsigned)iy < 15u) {
                    #pragma unroll
                    for (int kx = 0; kx < 3; ++kx) {
                        const int ix = x + kx - 1;
                        if ((unsigned)ix < 15u) acc += w[ky * 3 + kx] * f0[iy * 15 + ix];
                    }
                }
            }
            s_feat[ch * HWP + px] = acc > 0.0f ? acc : 0.0f;
        }
    } else if (px < HWP) {
        for (int ch = 0; ch < D; ++ch) s_feat[ch * HWP + px] = 0.0f;  // WMMA pad
    }
    __syncthreads();

    // ---------------- Stage 3: pointwise convs via WMMA f32 16x16x4 ----------------
    {
        const int lane = tid & 31;
        const int wave = tid >> 5;
        const int lm = lane & 15;   // M (for A) / N (for B,C,D)
        const int lh = lane >> 4;   // half-wave select

        // ---- value pointwise: 30 tiles (15 M x 2 N-halves); stride-8 => nh fixed per wave ----
        {
            const int nh = wave & 1;
            const int n  = nh * 16 + lm;               // value out-channel
            v2f wb[8];
            #pragma unroll
            for (int j = 0; j < 8; ++j) {              // preload B fragments (loop-invariant)
                const int ka = 4 * j + 2 * lh;
                wb[j].x = vpw_w[n * V + ka];
                wb[j].y = vpw_w[n * V + ka + 1];
            }
            const float bias = vpw_b[n];
            float s = 0.0f;
            for (int t = wave; t < 30; t += 8) {
                const int m0 = (t >> 1) * 16;
                v8f acc = {};
                #pragma unroll
                for (int j = 0; j < 8; ++j) {
                    const int ka = 4 * j + 2 * lh;
                    v2f a;
                    a.x = s_feat[(P + ka)     * HWP + m0 + lm];
                    a.y = s_feat[(P + ka + 1) * HWP + m0 + lm];
                    acc = __builtin_amdgcn_wmma_f32_16x16x4_f32(
                        false, a, false, wb[j], (short)0, acc, false, false);
                }
                #pragma unroll
                for (int r = 0; r < 8; ++r) {
                    const int p = m0 + r + 8 * lh;
                    float vv = acc[r] + bias;
                    vv = vv > 0.0f ? vv : 0.0f;
                    s += (p < HW) ? vv : 0.0f;         // branchless mask of pad pixels
                }
            }
            atomicAdd(&s_vsum[n], s);                  // single ds_add_f32 per lane
        }

        // ---- policy pointwise: 15 tiles ----
        {
            const int n = lm;                          // policy out-channel
            v2f wb[4];
            #pragma unroll
            for (int j = 0; j < 4; ++j) {
                const int ka = 4 * j + 2 * lh;
                wb[j].x = ppw_w[n * P + ka];
                wb[j].y = ppw_w[n * P + ka + 1];
            }
            const float bias = ppw_b[n];
            for (int t = wave; t < 15; t += 8) {
                const int m0 = t * 16;
                v8f acc = {};
                #pragma unroll
                for (int j = 0; j < 4; ++j) {
                    const int ka = 4 * j + 2 * lh;
                    v2f a;
                    a.x = s_feat[ka       * HWP + m0 + lm];
                    a.y = s_feat[(ka + 1) * HWP + m0 + lm];
                    acc = __builtin_amdgcn_wmma_f32_16x16x4_f32(
                        false, a, false, wb[j], (short)0, acc, false, false);
                }
                // lane's 8 D elements are contiguous pixels: two aligned float4 LDS stores.
                // pad pixels (>=225) land in the never-read tail of the 240-wide row.
                float4 s0, s1;
                s0.x = fmaxf(acc[0] + bias, 0.0f);
                s0.y = fmaxf(acc[1] + bias, 0.0f);
                s0.z = fmaxf(acc[2] + bias, 0.0f);
                s0.w = fmaxf(acc[3] + bias, 0.0f);
                s1.x = fmaxf(acc[4] + bias, 0.0f);
                s1.y = fmaxf(acc[5] + bias, 0.0f);
                s1.z = fmaxf(acc[6] + bias, 0.0f);
                s1.w = fmaxf(acc[7] + bias, 0.0f);
                float* dst = &s_pol1[n * HWP + m0 + 8 * lh];
                ((float4*)dst)[0] = s0;
                ((float4*)dst)[1] = s1;
            }
        }
    }
    __syncthreads();

    // ---------------- Stage 4a: policy depthwise 3x3 + ReLU + 1x1 (16->1) ----------------
    if (px < HW) {
        const int y = px / 15, x = px % 15;
        float o = 0.0f;
        for (int ch = 0; ch < P; ++ch) {
            float acc = pdw_b[ch];
            const float* w  = pdw_w + ch * 9;
            const float* f0 = s_pol1 + ch * HWP;
            #pragma unroll
            for (int ky = 0; ky < 3; ++ky) {
                const int iy = y + ky - 1;
                if ((unsigned)iy < 15u) {
                    #pragma unroll
                    for (int kx = 0; kx < 3; ++kx) {
                        const int ix = x + kx - 1;
                        if ((unsigned)ix < 15u) acc += w[ky * 3 + kx] * f0[iy * 15 + ix];
                    }
                }
            }
            acc = acc > 0.0f ? acc : 0.0f;
            o += pf_w[ch] * acc;
        }
        out_pol[(size_t)img * HW + px] = o;
    }

    // ---------------- Stage 4b: value MLP head (wave 0, shuffle broadcasts) ----------------
    if (tid < 32) {
        const int n = tid;
        float v = s_vsum[n] * (1.0f / 225.0f);   // global mean
        float h1 = vl1_b[n];
        #pragma unroll
        for (int k = 0; k < V; ++k) h1 += vl1_w[n * V + k] * __shfl(v, k, 32);
        h1 = h1 > 0.0f ? h1 : 0.0f;
        float h2 = vl2_b[n];
        #pragma unroll
        for (int k = 0; k < V; ++k) h2 += vl2_w[n * V + k] * __shfl(h1, k, 32);
        h2 = h2 > 0.0f ? h2 : 0.0f;
        const int j = n < 3 ? n : 0;             // keep all lanes active for shuffles
        float o = vf_b[j];
        #pragma unroll
        for (int k = 0; k < V; ++k) o += vf_w[j * V + k] * __shfl(h2, k, 32);
        if (n < 3) out_val[(size_t)img * 3 + n] = o;
    }
}

extern "C" void kernel_launch(void* const* d_in, const int* in_sizes, int n_in,
                              void* d_out, int out_size, void* d_ws, size_t ws_size,
                              hipStream_t stream) {
    const float* emb    = (const float*)d_in[0];
    const float* conv_w = (const float*)d_in[1];
    const float* conv_b = (const float*)d_in[2];
    const float* ppw_w  = (const float*)d_in[3];
    const float* ppw_b  = (const float*)d_in[4];
    const float* pdw_w  = (const float*)d_in[5];
    const float* pdw_b  = (const float*)d_in[6];
    const float* pf_w   = (const float*)d_in[7];
    const float* vpw_w  = (const float*)d_in[8];
    const float* vpw_b  = (const float*)d_in[9];
    const float* vl1_w  = (const float*)d_in[10];
    const float* vl1_b  = (const float*)d_in[11];
    const float* vl2_w  = (const float*)d_in[12];
    const float* vl2_b  = (const float*)d_in[13];
    const float* vf_w   = (const float*)d_in[14];
    const float* vf_b   = (const float*)d_in[15];
    const int* sparse   = (const int*)d_in[16];
    const int* board    = (const int*)d_in[17];

    float* out = (float*)d_out;
    float* out_val = out;                     // (B, 3) first
    float* out_pol = out + (size_t)BATCH * 3; // then (B, 1, 15, 15)

    patnet_fused<<<BATCH, 256, 0, stream>>>(
        emb, conv_w, conv_b, ppw_w, ppw_b, pdw_w, pdw_b, pf_w,
        vpw_w, vpw_b, vl1_w, vl1_b, vl2_w, vl2_b, vf_w, vf_b,
        sparse, board, out_val, out_pol);
}